// max_graph_node_features_39745627357562
// MI455X (gfx1250) — compile-verified
//
#include <hip/hip_runtime.h>
#include <math.h>

// segment_max over sorted contiguous segment ids.
//   x:    [N_NODES, 256] fp32, row-major
//   seg:  [N_NODES] int32, sorted, values 0..n_seg-1, every segment nonempty
//   out:  [n_seg, 256] fp32
//
// One 256-thread block per segment. 8-row (8KB) chunks staged global->LDS
// with gfx1250 async loads (ASYNCcnt), 3-deep ring buffer => 24KB in flight
// per block, max-reduced in VGPRs.

#define CHUNK_ROWS 8          // rows per chunk (2 per thread row-slot)
#define ROW_SLOTS  4          // tid>>6
#define F4_PER_ROW 64         // 256 floats / 4
#define NBUF       3          // pipeline depth

__device__ __forceinline__ int lower_bound_i32(const int* __restrict__ a,
                                               int n, int key) {
  int lo = 0, hi = n;
  while (lo < hi) {
    int mid = (lo + hi) >> 1;
    if (a[mid] < key) lo = mid + 1;
    else hi = mid;
  }
  return lo;
}

__global__ __launch_bounds__(256)
void max_graph_node_features_39745627357562_kernel(
    const float* __restrict__ x, const int* __restrict__ seg,
    float* __restrict__ out, int n_nodes) {
  // 3 x 8KB async ring buffer + 4KB cross-slot reduce = 28KB LDS
  __shared__ float4 buf[NBUF][CHUNK_ROWS][F4_PER_ROW];
  __shared__ float4 red[ROW_SLOTS][F4_PER_ROW];

  const int s   = blockIdx.x;          // segment id
  const int tid = threadIdx.x;
  const int q   = tid >> 6;            // row slot within chunk, 0..3
  const int c   = tid & 63;            // float4 column, 0..63

  // Segment bounds via binary search on the sorted splitter (uniform; the
  // compiler scalarizes these to s_load).
  const int start = lower_bound_i32(seg, n_nodes, s);
  const int end   = lower_bound_i32(seg, n_nodes, s + 1);
  const int rows  = end - start;                         // >= 1 by construction
  const int nchunks = (rows + CHUNK_ROWS - 1) / CHUNK_ROWS;

  const float4* __restrict__ x4 = (const float4*)x;

  // Stage chunk `it` (8 rows x 1KB) into buf[it%3]: two async b128 issues per
  // thread (rows q and q+4). Row indices are CLAMPED so every lane always
  // issues -> exact per-wave ASYNCcnt bookkeeping (2 ops/wave/chunk); tail
  // duplicates are guarded out at consumption.
  auto stage = [&](int it) {
    const int p  = it % NBUF;
    const int rb = start + it * CHUNK_ROWS;
    int r0 = rb + q;
    int r1 = rb + q + ROW_SLOTS;
    int rc0 = (r0 < end) ? r0 : (end - 1);
    int rc1 = (r1 < end) ? r1 : (end - 1);
    const float4* gp0 = x4 + (size_t)rc0 * F4_PER_ROW + c;
    const float4* gp1 = x4 + (size_t)rc1 * F4_PER_ROW + c;
    // LDS aperture: generic addr bits [31:0] == wave-relative LDS byte offset.
    unsigned la0 = (unsigned)(unsigned long long)(const void*)&buf[p][q][c];
    unsigned la1 =
        (unsigned)(unsigned long long)(const void*)&buf[p][q + ROW_SLOTS][c];
    asm volatile("global_load_async_to_lds_b128 %0, %1, off"
                 :: "v"(la0), "v"(gp0) : "memory");
    asm volatile("global_load_async_to_lds_b128 %0, %1, off"
                 :: "v"(la1), "v"(gp1) : "memory");
  };

  float4 acc = make_float4(-INFINITY, -INFINITY, -INFINITY, -INFINITY);

  stage(0);
  if (nchunks > 1) stage(1);

  for (int it = 0; it < nchunks; ++it) {
    if (it + 2 < nchunks) {
      stage(it + 2);
      // chunks it,it+1,it+2 in flight (6 ops); <=4 => chunk `it` complete
      asm volatile("s_wait_asynccnt 0x4" ::: "memory");
    } else if (it + 1 < nchunks) {
      // chunks it,it+1 in flight (4 ops); <=2 => chunk `it` complete
      asm volatile("s_wait_asynccnt 0x2" ::: "memory");
    } else {
      asm volatile("s_wait_asynccnt 0x0" ::: "memory");
    }
    __syncthreads();  // all waves' async writes for chunk `it` visible

    const int p  = it % NBUF;
    const int rb = start + it * CHUNK_ROWS;
    const int r0 = rb + q;
    const int r1 = rb + q + ROW_SLOTS;
    if (r0 < end) {
      float4 v = buf[p][q][c];                       // ds_load_b128
      acc.x = fmaxf(acc.x, v.x);
      acc.y = fmaxf(acc.y, v.y);
      acc.z = fmaxf(acc.z, v.z);
      acc.w = fmaxf(acc.w, v.w);
    }
    if (r1 < end) {
      float4 v = buf[p][q + ROW_SLOTS][c];           // ds_load_b128
      acc.x = fmaxf(acc.x, v.x);
      acc.y = fmaxf(acc.y, v.y);
      acc.z = fmaxf(acc.z, v.z);
      acc.w = fmaxf(acc.w, v.w);
    }
    __syncthreads();  // protect buf[it%3] against the chunk `it+3` overwrite
  }

  // Collapse the 4 row-slot partials.
  red[q][c] = acc;
  __syncthreads();
  if (tid < F4_PER_ROW) {
    float4 m = red[0][tid];
#pragma unroll
    for (int qq = 1; qq < ROW_SLOTS; ++qq) {
      float4 v = red[qq][tid];
      m.x = fmaxf(m.x, v.x);
      m.y = fmaxf(m.y, v.y);
      m.z = fmaxf(m.z, v.z);
      m.w = fmaxf(m.w, v.w);
    }
    ((float4*)out)[(size_t)s * F4_PER_ROW + tid] = m;  // global_store_b128
  }
}

extern "C" void kernel_launch(void* const* d_in, const int* in_sizes, int n_in,
                              void* d_out, int out_size, void* d_ws, size_t ws_size,
                              hipStream_t stream) {
  const float* x   = (const float*)d_in[0];
  const int*   seg = (const int*)d_in[1];
  float*       out = (float*)d_out;

  const int n_nodes = in_sizes[1];       // splitter element count
  const int n_seg   = out_size / 256;    // D_FEAT = 256

  dim3 grid(n_seg), block(256);
  hipLaunchKernelGGL(max_graph_node_features_39745627357562_kernel,
                     grid, block, 0, stream, x, seg, out, n_nodes);
}